// SelfAttention3D_584115552856
// MI455X (gfx1250) — compile-verified
//
#include <hip/hip_runtime.h>
#include <hip/hip_bf16.h>

// SelfAttention3D over the W axis, fused single pass.
// B=2, C=64, C/8=8, D=H=W=64. Memory-bound (268MB @ 23.3TB/s ~= 11.5us),
// so we use full-precision FP32 WMMA (v_wmma_f32_16x16x4_f32): exact vs the
// fp32 reference and still under the HBM roofline. Staging uses the CDNA5
// async global->LDS path (ASYNCcnt) instead of VGPR round-trips.

typedef __attribute__((ext_vector_type(2))) float v2f;
typedef __attribute__((ext_vector_type(8))) float v8f;

#define CC   64
#define WW   64
#define DHW  (64 * 64 * 64)
#define SX   68   // Xs[c][w] stride: 68*4B = 272B rows -> 16B-aligned for B128 async
#define SV   65   // Vs[w][c'] / Bv[c][c'] / Es[w][u] stride
#define SE   65
#define SB   17   // Bqk[c][j] stride
#define SQ   17   // QKs[w][j] stride

// ---- CDNA5 async global->LDS copies (tracked by ASYNCcnt) ----
// VDST = 32-bit wave-relative LDS byte address (low 32 bits of flat shared ptr,
// per the LDS aperture rule addr[31:0]); VADDR = 64-bit global address.
__device__ __forceinline__ void async_copy_b128(const void* g, void* lds) {
    asm volatile("global_load_async_to_lds_b128 %0, %1, off"
                 :: "v"((unsigned)(uintptr_t)lds), "v"(g) : "memory");
}
__device__ __forceinline__ void async_copy_b32(const void* g, void* lds) {
    asm volatile("global_load_async_to_lds_b32 %0, %1, off"
                 :: "v"((unsigned)(uintptr_t)lds), "v"(g) : "memory");
}
__device__ __forceinline__ void wait_async0() {
#if __has_builtin(__builtin_amdgcn_s_wait_asynccnt)
    __builtin_amdgcn_s_wait_asynccnt(0);
#else
    asm volatile("s_wait_asynccnt 0x0" ::: "memory");
#endif
}

__global__ __launch_bounds__(256) void
SelfAttention3D_584115552856_kernel(const float* __restrict__ x,
                                    const float* __restrict__ wq,
                                    const float* __restrict__ bq,
                                    const float* __restrict__ wk,
                                    const float* __restrict__ bk,
                                    const float* __restrict__ wv,
                                    const float* __restrict__ bv,
                                    const float* __restrict__ gamma,
                                    float* __restrict__ out)
{
    // Static LDS: 4352+4160+4160+1088+1088+16+64 floats = ~58.3 KB.
    __shared__ float Xs[CC * SX];     // x row, Xs[c][w]           (live whole kernel)
    __shared__ float Vs[WW * SV];     // V[w][c']                  (phase1 -> phase4)
    __shared__ float PoolB[CC * SV];  // phase1: Wv^T[c][c']; phase2+: E/attn[w][u]
    __shared__ float Bqk[CC * SB];    // [Wq;Wk]^T[c][j]           (phase1 only)
    __shared__ float QKs[WW * SQ];    // q|k per position          (phase1 -> phase2)
    __shared__ float bqk_s[16];
    __shared__ float bv_s[64];

    const int t    = threadIdx.x;
    const int lane = t & 31;
    // Force SGPR wave id -> scalar branches, EXEC stays all-ones at every WMMA.
    const int wave = __builtin_amdgcn_readfirstlane(t >> 5);

    const int row = blockIdx.x;            // b*(D*H) + d*H + h
    const int b   = row >> 12;
    const int dh  = row & 4095;
    const size_t base = (size_t)b * (size_t)(CC * DHW) + (size_t)dh * WW;

    // ---------------- Phase 0: async-stage operands into LDS ----------------
    {   // x row: 64 channels x 64 w (256B contiguous per channel), 16B per lane
        const int c0 = t >> 4;             // 0..15
        const int w4 = (t & 15) << 2;
        for (int it = 0; it < 4; ++it) {
            const int c = c0 + (it << 4);
            async_copy_b128(x + base + (size_t)c * DHW + w4, &Xs[c * SX + w4]);
        }
    }
    for (int idx = t; idx < 4096; idx += 256) {   // Wv^T transposed during the copy
        const int cp = idx >> 6, c = idx & 63;
        async_copy_b32(wv + idx, &PoolB[c * SV + cp]);     // wv[cp*64 + c]
    }
    for (int idx = t; idx < 1024; idx += 256) {   // [Wq;Wk]^T
        const int o = idx >> 6, c = idx & 63;
        const float* src = (o < 8) ? (wq + o * 64 + c) : (wk + (o - 8) * 64 + c);
        async_copy_b32(src, &Bqk[c * SB + o]);
    }
    if (t < 16)       bqk_s[t]     = (t < 8) ? bq[t] : bk[t - 8];
    else if (t < 80)  bv_s[t - 16] = bv[t - 16];
    wait_async0();
    __syncthreads();

    const int m16   = lane & 15;            // M (or N) index within tile
    const int kOff  = (lane >> 4) << 1;     // K pair selected by lane half
    const int half8 = (lane >> 4) << 3;     // +8 rows for upper lane half (C/D layout)

    // ---------------- Phase 1: V = X^T * Wv^T  and  QK = X^T * [Wq;Wk]^T ----------------
    for (int job = wave; job < 20; job += 8) {
        v8f acc = {};
        if (job < 16) {                                    // V tile (wTile, cTile)
            const int wBase = (job >> 2) << 4;
            const int cBase = (job & 3) << 4;
            #pragma unroll
            for (int k = 0; k < 64; k += 4) {
                v2f a, bb;
                a.x  = Xs[(k + kOff)     * SX + wBase + m16];
                a.y  = Xs[(k + kOff + 1) * SX + wBase + m16];
                bb.x = PoolB[(k + kOff)     * SV + cBase + m16];
                bb.y = PoolB[(k + kOff + 1) * SV + cBase + m16];
                acc = __builtin_amdgcn_wmma_f32_16x16x4_f32(false, a, false, bb,
                                                            (short)0, acc, false, false);
            }
            #pragma unroll
            for (int r = 0; r < 8; ++r)
                Vs[(wBase + r + half8) * SV + cBase + m16] = acc[r] + bv_s[cBase + m16];
        } else {                                           // QK tile (wTile, 16 cols = q|k)
            const int wBase = (job - 16) << 4;
            #pragma unroll
            for (int k = 0; k < 64; k += 4) {
                v2f a, bb;
                a.x  = Xs[(k + kOff)     * SX + wBase + m16];
                a.y  = Xs[(k + kOff + 1) * SX + wBase + m16];
                bb.x = Bqk[(k + kOff)     * SB + m16];
                bb.y = Bqk[(k + kOff + 1) * SB + m16];
                acc = __builtin_amdgcn_wmma_f32_16x16x4_f32(false, a, false, bb,
                                                            (short)0, acc, false, false);
            }
            #pragma unroll
            for (int r = 0; r < 8; ++r)
                QKs[(wBase + r + half8) * SQ + m16] = acc[r] + bqk_s[m16];
        }
    }
    __syncthreads();   // QKs/Vs ready; PoolB(Wv^T) now dead -> reuse as E

    // ---------------- Phase 2: E = Q * K^T  (K-dim = 8) ----------------
    for (int job = wave; job < 16; job += 8) {
        const int wBase = (job >> 2) << 4;
        const int uBase = (job & 3) << 4;
        v8f acc = {};
        #pragma unroll
        for (int k = 0; k < 8; k += 4) {
            v2f a, bb;
            a.x  = QKs[(wBase + m16) * SQ + k + kOff];
            a.y  = QKs[(wBase + m16) * SQ + k + kOff + 1];
            bb.x = QKs[(uBase + m16) * SQ + 8 + k + kOff];      // K^T[j][u] = QKs[u][8+j]
            bb.y = QKs[(uBase + m16) * SQ + 8 + k + kOff + 1];
            acc = __builtin_amdgcn_wmma_f32_16x16x4_f32(false, a, false, bb,
                                                        (short)0, acc, false, false);
        }
        #pragma unroll
        for (int r = 0; r < 8; ++r)
            PoolB[(wBase + r + half8) * SE + uBase + m16] = acc[r];
    }
    __syncthreads();

    // ---------------- Phase 3: softmax over u (4 threads per row) ----------------
    {
        const int w  = t >> 2;
        const int q4 = t & 3;
        float* rowp = &PoolB[w * SE + (q4 << 4)];
        float e[16];
        float mx = -3.402823466e+38f;
        #pragma unroll
        for (int i = 0; i < 16; ++i) { e[i] = rowp[i]; mx = fmaxf(mx, e[i]); }
        mx = fmaxf(mx, __shfl_xor(mx, 1, 32));
        mx = fmaxf(mx, __shfl_xor(mx, 2, 32));
        float s = 0.f;
        #pragma unroll
        for (int i = 0; i < 16; ++i) { e[i] = __expf(e[i] - mx); s += e[i]; }
        s += __shfl_xor(s, 1, 32);
        s += __shfl_xor(s, 2, 32);
        const float inv = 1.0f / s;
        #pragma unroll
        for (int i = 0; i < 16; ++i) rowp[i] = e[i] * inv;
    }
    __syncthreads();

    // ---------------- Phase 4: O^T = V^T * attn^T, fused epilogue ----------------
    // D-fragment rows = c', cols = w  ->  coalesced 64B global stores, residual from Xs.
    const float g = gamma[0];
    for (int job = wave; job < 16; job += 8) {
        const int cBase = (job >> 2) << 4;
        const int wBase = (job & 3) << 4;
        v8f acc = {};
        #pragma unroll
        for (int k = 0; k < 64; k += 4) {
            v2f a, bb;
            a.x  = Vs[(k + kOff)     * SV + cBase + m16];       // V^T[c'][u] = Vs[u][c']
            a.y  = Vs[(k + kOff + 1) * SV + cBase + m16];
            bb.x = PoolB[(wBase + m16) * SE + k + kOff];        // attn^T[u][w] = attn[w][u]
            bb.y = PoolB[(wBase + m16) * SE + k + kOff + 1];
            acc = __builtin_amdgcn_wmma_f32_16x16x4_f32(false, a, false, bb,
                                                        (short)0, acc, false, false);
        }
        #pragma unroll
        for (int r = 0; r < 8; ++r) {
            const int cp = cBase + r + half8;
            out[base + (size_t)cp * DHW + wBase + m16] =
                g * acc[r] + Xs[cp * SX + wBase + m16];
        }
    }
}

extern "C" void kernel_launch(void* const* d_in, const int* in_sizes, int n_in,
                              void* d_out, int out_size, void* d_ws, size_t ws_size,
                              hipStream_t stream) {
    const float* x  = (const float*)d_in[0];
    const float* wq = (const float*)d_in[1];
    const float* bq = (const float*)d_in[2];
    const float* wk = (const float*)d_in[3];
    const float* bk = (const float*)d_in[4];
    const float* wv = (const float*)d_in[5];
    const float* bv = (const float*)d_in[6];
    const float* gm = (const float*)d_in[7];
    float* out = (float*)d_out;

    dim3 grid(2 * 64 * 64);   // one block per (b,d,h) attention row
    dim3 block(256);          // 8 wave32
    hipLaunchKernelGGL(SelfAttention3D_584115552856_kernel, grid, block, 0, stream,
                       x, wq, bq, wk, bk, wv, bv, gm, out);
}